// HungarianLoss_27049704030883
// MI455X (gfx1250) — compile-verified
//
#include <hip/hip_runtime.h>
#include <cstdint>

#define LAP_N  256
#define LAP_NT 256

// gfx1250 async global->LDS staging (probe-confirmed present on this toolchain,
// signature: (v4i addrspace(1)*, v4i addrspace(3)*, imm offset, imm cpol)).
#if defined(__has_builtin)
#  if __has_builtin(__builtin_amdgcn_global_load_async_to_lds_b128) && \
      __has_builtin(__builtin_amdgcn_s_wait_asynccnt)
#    define LAP_ASYNC_STAGE 1
#  endif
#endif

typedef int lap_v4i __attribute__((ext_vector_type(4)));
typedef lap_v4i __attribute__((address_space(1)))* lap_gptr;   // global v4i*
typedef lap_v4i __attribute__((address_space(3)))* lap_lptr;   // LDS v4i*

__global__ __launch_bounds__(LAP_NT, 1)
void lap_solve_kernel(const float* __restrict__ D, float* __restrict__ partial)
{
    // 256 KB cost matrix + ~6.5 KB working state: needs CDNA5's 320 KB LDS/WGP.
    __shared__ float Cs[LAP_N * LAP_N];
    __shared__ float spc_s[LAP_N];         // shortest path cost per column (LDS mirror)
    __shared__ int   path_s[LAP_N];        // predecessor row per column
    __shared__ float u_s[LAP_N];           // row duals
    __shared__ int   col4row_s[LAP_N];
    __shared__ int   row4col_s[LAP_N];
    __shared__ int   vis_s[LAP_N];         // visited-row list (SR)
    __shared__ unsigned long long redpk[2][8];  // double-buffered (val,idx) partials
    __shared__ float ssum[8];

    const int b    = blockIdx.x;
    const int tid  = threadIdx.x;          // thread tid owns column tid
    const int lane = tid & 31;
    const int wv   = tid >> 5;
    const float INF = __builtin_inff();

    const float* Db = D + (size_t)b * LAP_N * LAP_N;

    // ---- stage this instance's cost matrix into LDS (async data mover path) ----
#if defined(LAP_ASYNC_STAGE)
    for (int k = tid; k < LAP_N * LAP_N / 4; k += LAP_NT) {
        const float4* gp = ((const float4*)Db) + k;
        float4*       lp = ((float4*)Cs) + k;
        __builtin_amdgcn_global_load_async_to_lds_b128(
            (lap_gptr)(void*)gp, (lap_lptr)(void*)lp, 0, 0);
    }
    __builtin_amdgcn_s_wait_asynccnt(0);
#else
    for (int k = tid; k < LAP_N * LAP_N / 4; k += LAP_NT) {
        ((float4*)Cs)[k] = ((const float4*)Db)[k];
    }
#endif

    u_s[tid]       = 0.0f;
    col4row_s[tid] = -1;
    row4col_s[tid] = -1;
    float v_reg = 0.0f;                    // column dual, register-resident
    __syncthreads();

    for (int cur_row = 0; cur_row < LAP_N; ++cur_row) {
        // per-augmentation reset (owner-local writes; no barrier needed — the
        // post-augmentation barrier of the previous round separates all readers)
        float spc = INF;                   // register copy of spc_s[tid]
        bool  sc  = false;                 // column-visited flag (SC)
        spc_s[tid]  = INF;
        path_s[tid] = -1;
        float minVal = 0.0f;
        int   i     = cur_row;             // current row (uniform)
        int   sink  = -1;
        int   nvis  = 0;
        int   par   = 0;                   // reduction buffer parity

        // ---- Dijkstra-style shortest augmenting path: ONE barrier per step ----
        while (sink < 0) {
            if (tid == 0) vis_s[nvis] = i; // SR.add(i); fresh slot each step
            nvis++;

            const float ui = u_s[i];                          // LDS broadcast
            const float r  = minVal + Cs[i * LAP_N + tid] - ui - v_reg;
            if (!sc && r < spc) { spc = r; spc_s[tid] = r; path_s[tid] = i; }

            // argmin over masked spc, lowest-index tie-break (matches jnp.argmin)
            float bv = sc ? INF : spc;
            int   bj = tid;
            #pragma unroll
            for (int off = 16; off > 0; off >>= 1) {
                const float ov = __shfl_down(bv, off, 32);
                const int   oj = __shfl_down(bj, off, 32);
                if (ov < bv || (ov == bv && oj < bj)) { bv = ov; bj = oj; }
            }
            if (lane == 0) {
                redpk[par][wv] = ((unsigned long long)__float_as_uint(bv) << 32)
                               | (unsigned int)bj;
            }
            __syncthreads();
            // every thread combines the 8 wave partials -> uniform result
            bv = INF; bj = LAP_N;
            #pragma unroll
            for (int w = 0; w < 8; ++w) {
                const unsigned long long pk = redpk[par][w];
                const float ov = __uint_as_float((unsigned int)(pk >> 32));
                const int   oj = (int)(unsigned int)(pk & 0xffffffffu);
                if (ov < bv || (ov == bv && oj < bj)) { bv = ov; bj = oj; }
            }
            minVal = bv;
            const int j = bj;
            if (tid == j) sc = true;                // SC.add(j), owner-local
            const int r4c = row4col_s[j];           // LDS broadcast (loop-invariant array)
            if (r4c < 0) sink = j; else i = r4c;
            par ^= 1;                               // double-buffer: no end-of-step barrier
        }

        // ---- dual variable update (scipy-identical), parallel over visited rows ----
        // All spc_s/vis_s writes happened before the final step's barrier.
        if (tid == 0) {
            u_s[cur_row] += minVal;                               // vis_s[0] == cur_row
        } else if (tid < nvis) {
            const int iv = vis_s[tid];
            u_s[iv] += minVal - spc_s[col4row_s[iv]];
        }
        if (sc) v_reg += spc - minVal;
        __syncthreads();   // dual reads of col4row_s must finish before augmentation writes

        // ---- augment along path: serial chain on thread 0 ----
        if (tid == 0) {
            int j = sink;
            for (;;) {
                const int ii = path_s[j];
                row4col_s[j] = ii;
                const int jn = col4row_s[ii];
                col4row_s[ii] = j;
                if (ii == cur_row) break;
                j = jn;
            }
        }
        __syncthreads();   // publish matching + protect next round's resets
    }

    // ---- per-instance mean of matched costs ----
    float m = Cs[tid * LAP_N + col4row_s[tid]];
    #pragma unroll
    for (int off = 16; off > 0; off >>= 1) m += __shfl_down(m, off, 32);
    if (lane == 0) ssum[wv] = m;
    __syncthreads();
    if (tid == 0) {
        float t = 0.0f;
        #pragma unroll
        for (int w = 0; w < 8; ++w) t += ssum[w];
        partial[b] = t * (1.0f / LAP_N);
    }
}

// Deterministic fixed-order final reduction: mean over batch.
__global__ __launch_bounds__(256)
void lap_reduce_kernel(const float* __restrict__ partial, float* __restrict__ out, int B)
{
    __shared__ float ss[8];
    const int tid  = threadIdx.x;
    const int lane = tid & 31;
    const int wv   = tid >> 5;
    float s = 0.0f;
    for (int k = tid; k < B; k += 256) s += partial[k];
    #pragma unroll
    for (int off = 16; off > 0; off >>= 1) s += __shfl_down(s, off, 32);
    if (lane == 0) ss[wv] = s;
    __syncthreads();
    if (tid == 0) {
        float t = 0.0f;
        for (int w = 0; w < 8; ++w) t += ss[w];
        out[0] = t / (float)B;
    }
}

extern "C" void kernel_launch(void* const* d_in, const int* in_sizes, int n_in,
                              void* d_out, int out_size, void* d_ws, size_t ws_size,
                              hipStream_t stream)
{
    (void)n_in; (void)out_size; (void)ws_size;
    const float* D = (const float*)d_in[0];
    const int B = in_sizes[0] / (LAP_N * LAP_N);   // 256 instances
    float* partial = (float*)d_ws;                 // B floats of scratch

    lap_solve_kernel<<<dim3(B), dim3(LAP_NT), 0, stream>>>(D, partial);
    lap_reduce_kernel<<<dim3(1), dim3(256), 0, stream>>>(partial, (float*)d_out, B);
}